// DTW_SPRING_NPATH_38448547233959
// MI455X (gfx1250) — compile-verified
//
#include <hip/hip_runtime.h>
#include <hip/hip_bf16.h>
#include <stdint.h>

// Problem constants (from reference)
constexpr int KERNEL_LEN = 128;
constexpr int INPUT_LEN  = 1000000;
constexpr int L          = INPUT_LEN + KERNEL_LEN;   // 1000128, divisible by 4
constexpr int PATH_NUM   = 10;
constexpr int NVEC       = L / 4;                    // 250032 int4 per row
constexpr int TPB        = 256;                      // 8 waves (wave32)

#if __has_builtin(__builtin_amdgcn_global_load_async_to_lds_b128) && \
    __has_builtin(__builtin_amdgcn_s_wait_asynccnt)
#define USE_ASYNC_LDS 1
#else
#define USE_ASYNC_LDS 0
#endif

typedef __attribute__((ext_vector_type(4))) int v4i;
typedef __attribute__((address_space(1))) v4i* gv4i_p;   // global (AS1) v4i*
typedef __attribute__((address_space(3))) v4i* sv4i_p;   // LDS (AS3) v4i*

__global__ __launch_bounds__(TPB) void dtw_spring_partial(
    const float* __restrict__ kern,
    const float* __restrict__ x,
    const int*   __restrict__ path_k,
    const int*   __restrict__ path_i,
    const int*   __restrict__ path_length,
    const int*   __restrict__ path_num,
    float*       __restrict__ partials,
    int blocks_per_path)
{
    __shared__ float k_lds[KERNEL_LEN];
    __shared__ float wsum[TPB / 32];

    const int tid = threadIdx.x;
    const int p   = blockIdx.y;          // path 0..9
    const int b   = blockIdx.x;          // chunk within path

    // ---- Stage kernel[] (512B) into LDS via CDNA5 async LDS DMA ----
#if USE_ASYNC_LDS
    if (tid < 32) {
        __builtin_amdgcn_global_load_async_to_lds_b128(
            (gv4i_p)(kern + tid * 4),        // 16B per lane, 32 lanes = 512B
            (sv4i_p)(&k_lds[tid * 4]),
            /*offset=*/0, /*cpol=*/0);
    }
    __builtin_amdgcn_s_wait_asynccnt(0);
#else
    if (tid < KERNEL_LEN) k_lds[tid] = kern[tid];
#endif
    __syncthreads();

    int n = *path_num;
    if (n > PATH_NUM) n = PATH_NUM;

    float acc = 0.0f;

    if (p < n) {
        const int len  = path_length[p];          // elements to include
        const int lenv = (len + 3) >> 2;          // int4 count touching valid range

        const int chunk = (NVEC + blocks_per_path - 1) / blocks_per_path;
        int start = b * chunk;
        int end   = start + chunk;
        if (end > NVEC) end = NVEC;
        if (end > lenv) end = lenv;               // whole-block early exit if start >= end

        const int4* __restrict__ pk4 =
            reinterpret_cast<const int4*>(path_k + (size_t)p * L);
        const int4* __restrict__ pi4 =
            reinterpret_cast<const int4*>(path_i + (size_t)p * L);

        for (int v = start + tid; v < end; v += TPB) {
            const int4 kk = pk4[v];               // global_load_b128
            const int4 ii = pi4[v];               // global_load_b128
            const int  t0 = v << 2;

            float d0 = k_lds[kk.x] - x[ii.x];
            float d1 = k_lds[kk.y] - x[ii.y];
            float d2 = k_lds[kk.z] - x[ii.z];
            float d3 = k_lds[kk.w] - x[ii.w];

            acc += (t0 + 0 < len) ? d0 * d0 : 0.0f;
            acc += (t0 + 1 < len) ? d1 * d1 : 0.0f;
            acc += (t0 + 2 < len) ? d2 * d2 : 0.0f;
            acc += (t0 + 3 < len) ? d3 * d3 : 0.0f;
        }
    }

    // ---- wave32 butterfly reduction, then cross-wave via LDS ----
    #pragma unroll
    for (int off = 16; off > 0; off >>= 1)
        acc += __shfl_xor(acc, off, 32);

    const int wave = tid >> 5;
    const int lane = tid & 31;
    if (lane == 0) wsum[wave] = acc;
    __syncthreads();

    if (tid == 0) {
        float total = 0.0f;
        #pragma unroll
        for (int w = 0; w < TPB / 32; ++w) total += wsum[w];
        partials[(size_t)p * blocks_per_path + b] = total;  // every block writes
    }
}

__global__ __launch_bounds__(TPB) void dtw_spring_reduce(
    const float* __restrict__ partials,
    const int*   __restrict__ path_num,
    float*       __restrict__ out,
    int blocks_per_path)
{
    __shared__ float s[TPB];
    const int p   = blockIdx.x;      // 10 blocks
    const int tid = threadIdx.x;

    // Fixed-order strided accumulate -> deterministic
    float a = 0.0f;
    for (int i = tid; i < blocks_per_path; i += TPB)
        a += partials[(size_t)p * blocks_per_path + i];
    s[tid] = a;
    __syncthreads();

    #pragma unroll
    for (int off = TPB / 2; off > 0; off >>= 1) {
        if (tid < off) s[tid] += s[tid + off];
        __syncthreads();
    }

    if (tid == 0) {
        int n = *path_num;
        if (n > PATH_NUM) n = PATH_NUM;
        out[p] = (p < n) ? s[0] : 0.0f;
    }
}

extern "C" void kernel_launch(void* const* d_in, const int* in_sizes, int n_in,
                              void* d_out, int out_size, void* d_ws, size_t ws_size,
                              hipStream_t stream)
{
    const float* kern     = (const float*)d_in[0];
    const float* x        = (const float*)d_in[1];
    const int*   path_k   = (const int*)d_in[2];
    const int*   path_i   = (const int*)d_in[3];
    const int*   path_len = (const int*)d_in[4];
    const int*   path_num = (const int*)d_in[5];
    float*       out      = (float*)d_out;
    float*       partials = (float*)d_ws;

    // Shrink blocks-per-path until partials fit in workspace.
    int bpp = 512;
    while ((size_t)bpp * PATH_NUM * sizeof(float) > ws_size && bpp > 1)
        bpp >>= 1;

    dim3 grid(bpp, PATH_NUM);
    dtw_spring_partial<<<grid, TPB, 0, stream>>>(
        kern, x, path_k, path_i, path_len, path_num, partials, bpp);

    dtw_spring_reduce<<<PATH_NUM, TPB, 0, stream>>>(partials, path_num, out, bpp);
}